// _AnchorLayer_59433757442289
// MI455X (gfx1250) — compile-verified
//
#include <hip/hip_runtime.h>
#include <hip/hip_bf16.h>
#include <stdint.h>

// ---------------------------------------------------------------------------
// Anchor-target assignment (RPN) for MI455X / gfx1250.
//
// Roofline: ~12 MB HBM traffic -> ~0.5us @ 23.3 TB/s; ~0.25 GFLOP of min/max
// VALU work. Launch-overhead bound, so the pipeline is fused to 4 kernels.
// No GEMM structure => WMMA inapplicable. CDNA5-specific paths used:
//   * global_load_async_to_lds_b128 + s_wait_asynccnt (ASYNCcnt data mover)
//     to stage the 1KB gt-box table into LDS per workgroup
//   * ds_max_u64 LDS atomics + global_atomic_max_u64 packed argmax
//   * non-temporal (TH=NT) b128 loads/stores for read-once/write-once streams
//   * wave32-safe code throughout
// ---------------------------------------------------------------------------

typedef __attribute__((ext_vector_type(4))) float v4f;   // NT builtins need a
                                                         // native clang vector

#define NA      15
#define HW      128
#define KANCH   (NA * HW * HW)      // 245760, divisible by 256 (960 blocks)
#define G       64
#define POS_T   0.7f
#define NEG_T   0.3f
#define NUM_FG  85                  // 256 // 3
#define NUM_BG  170                 // 256 * 2 // 3
#define SEED_BG 0x1234ABCDu
#define SEED_FG 0x9E3779B9u

__device__ __forceinline__ unsigned hash_u32(unsigned x, unsigned seed) {
    x ^= seed;
    x *= 0x9E3779B1u; x ^= x >> 16;
    x *= 0x85EBCA6Bu; x ^= x >> 13;
    x *= 0xC2B2AE35u; x ^= x >> 16;
    return x;
}

// Stage the 64 gt boxes (64 * 16B = 1KB) into LDS via the CDNA5 async
// global->LDS path (tracked with ASYNCcnt), then make visible via barrier.
__device__ __forceinline__ void stage_gt_async(const float* __restrict__ gt,
                                               float* s_gt, int tid) {
    if (tid < G) {
        unsigned lds_off = (unsigned)(uintptr_t)(&s_gt[tid * 4]);
        const float* src = gt + tid * 4;
        asm volatile("global_load_async_to_lds_b128 %0, %1, off"
                     :: "v"(lds_off), "v"(src)
                     : "memory");
    }
    asm volatile("s_wait_asynccnt 0" ::: "memory");
    __syncthreads();
}

// ---------------------------------------------------------------------------
// K0: init per-gt argmax packs and zero histograms.
// Pack layout: [63:32] = float bits of overlap (>0 so bit-monotonic),
//              [31:0]  = ~anchor_idx  (max() tie-breaks to smallest idx,
//                        matching jnp.argmax first-occurrence semantics).
__global__ void k_init(unsigned long long* __restrict__ gt_pack,
                       unsigned* __restrict__ hist) {
    int t = threadIdx.x;
    if (t < G)
        gt_pack[t] = ((unsigned long long)__float_as_uint(1e-10f) << 32)
                   | 0xFFFFFFFFull;          // overlap 1e-10 at anchor 0
    if (t < 512) hist[t] = 0u;
}

// ---------------------------------------------------------------------------
// K1: fused per-anchor IoU pass + preliminary-label histogram.
__global__ void k_iou(const float* __restrict__ anchors,
                      const float* __restrict__ gt,
                      int* __restrict__ labels,
                      int* __restrict__ arg_ov,
                      unsigned char* __restrict__ validb,
                      unsigned long long* __restrict__ gt_pack,
                      unsigned* __restrict__ hist) {
    __shared__ __align__(16) float s_gt[G * 4];
    __shared__ unsigned long long s_gtmax[G];
    __shared__ unsigned sh[512];             // bg hist [0,256), fg hist [256,512)

    const int tid = threadIdx.x;
    const int k   = blockIdx.x * blockDim.x + tid;

    if (tid < G) s_gtmax[tid] = 0ull;
    sh[tid] = 0u; sh[tid + 256] = 0u;
    stage_gt_async(gt, s_gt, tid);           // includes __syncthreads()

    // read-once stream: non-temporal b128 load
    const v4f a  = __builtin_nontemporal_load((const v4f*)anchors + k);
    const float ax = a.x, ay = a.y, aw = a.z, ah = a.w;
    const float Lm1 = (float)(HW - 1);
    const bool valid = (ax >= 0.f) & (ay >= 0.f) & (aw >= 0.f) & (ah >= 0.f) &
                       (ax <= Lm1) & (ay <= Lm1) &
                       (ax + aw - 1.f <= Lm1) & (ay + ah - 1.f <= Lm1);

    const float ax2 = ax + aw, ay2 = ay + ah, aArea = aw * ah;
    float best = -1.f;
    int bestg = 0;

    #pragma unroll 4
    for (int g = 0; g < G; ++g) {
        const v4f gb = ((const v4f*)s_gt)[g];         // ds_load_b128
        float iw = fminf(ax2, gb.x + gb.z) - fmaxf(ax, gb.x);
        float ih = fminf(ay2, gb.y + gb.w) - fmaxf(ay, gb.y);
        iw = fmaxf(iw, 0.f);
        ih = fmaxf(ih, 0.f);
        const float inter = iw * ih;
        const float uni   = aArea + gb.z * gb.w - inter;
        const float iou   = inter / uni;
        float ov = valid ? iou : 0.f;
        ov = (ov == 0.f) ? 1e-10f : ov;
        if (ov > best) { best = ov; bestg = g; }   // strict > => first max wins
        if (ov > 1e-10f) {                          // ~2 hits/thread average
            const unsigned long long pack =
                ((unsigned long long)__float_as_uint(ov) << 32) |
                (unsigned long long)(0xFFFFFFFFu - (unsigned)k);
            atomicMax(&s_gtmax[g], pack);           // ds_max_u64
        }
    }

    int lab = -1;
    if (valid) lab = (best >= POS_T) ? 1 : ((best <= NEG_T) ? 0 : -1);
    labels[k]  = lab;
    arg_ov[k]  = bestg;
    validb[k]  = valid ? 1 : 0;

    // preliminary subsample histograms (corrected for gt_arg flips in K2)
    if (lab == 0)      atomicAdd(&sh[hash_u32((unsigned)k, SEED_BG) >> 24], 1u);
    else if (lab == 1) atomicAdd(&sh[256 + (hash_u32((unsigned)k, SEED_FG) >> 24)], 1u);

    __syncthreads();
    if (tid < G) {
        const unsigned long long v = s_gtmax[tid];
        if (v) atomicMax(&gt_pack[tid], v);         // global_atomic_max_u64
    }
    if (sh[tid])       atomicAdd(&hist[tid],       sh[tid]);
    if (sh[tid + 256]) atomicAdd(&hist[tid + 256], sh[tid + 256]);
}

// ---------------------------------------------------------------------------
// K2 (single block): apply gt_arg label forcing, correct the histograms for
// the flipped anchors, then scan for subsample thresholds.
__global__ void k_fix_thresh(const unsigned long long* __restrict__ gt_pack,
                             const unsigned char* __restrict__ validb,
                             int* __restrict__ labels,
                             const unsigned* __restrict__ hist,
                             unsigned* __restrict__ ctrl) {
    __shared__ int s_delta[512];
    const int tid = threadIdx.x;                 // 256 threads
    s_delta[tid] = 0; s_delta[tid + 256] = 0;
    __syncthreads();

    if (tid < G) {
        const unsigned kk =
            0xFFFFFFFFu - (unsigned)(gt_pack[tid] & 0xFFFFFFFFull);
        if (kk < (unsigned)KANCH && validb[kk]) {
            // atomicExch so a doubly-referenced anchor is corrected once
            const int old = atomicExch(&labels[kk], 1);
            if (old == 0)
                atomicSub(&s_delta[hash_u32(kk, SEED_BG) >> 24], 1);
            if (old != 1)
                atomicAdd(&s_delta[256 + (hash_u32(kk, SEED_FG) >> 24)], 1);
        }
    }
    __syncthreads();

    if (tid == 0) {
        for (int which = 0; which < 2; ++which) {
            const unsigned* h = hist + which * 256;
            const int*      d = s_delta + which * 256;
            const unsigned limit = (which == 0) ? NUM_BG : NUM_FG;
            unsigned cum = 0, thr = 0, quota = 0xFFFFFFFFu;
            for (int b = 255; b >= 0; --b) {
                const unsigned cnt = (unsigned)((int)h[b] + d[b]);
                const unsigned nc  = cum + cnt;
                if (nc >= limit) { thr = (unsigned)b; quota = limit - cum; break; }
                cum = nc;
            }
            ctrl[2 * which + 0] = thr;       // bins > thr always survive
            ctrl[2 * which + 1] = quota;     // survivors allowed in bin thr
        }
        ctrl[4] = 0u;                        // bg boundary counter
        ctrl[5] = 0u;                        // fg boundary counter
    }
}

// ---------------------------------------------------------------------------
// K3: apply subsampling, emit final int32 labels + float4 regression targets.
__global__ void k_final(const float* __restrict__ anchors,
                        const float* __restrict__ gt,
                        const int* __restrict__ arg_ov,
                        int* __restrict__ labels,
                        float* __restrict__ tgt,
                        unsigned* __restrict__ ctrl) {
    __shared__ __align__(16) float s_gt[G * 4];
    const int tid = threadIdx.x;
    const int k   = blockIdx.x * 256 + tid;
    stage_gt_async(gt, s_gt, tid);

    int lab = labels[k];
    if (lab == 0) {
        const unsigned bin = hash_u32((unsigned)k, SEED_BG) >> 24;
        if (bin < ctrl[0]) lab = -1;
        else if (bin == ctrl[0] && atomicAdd(&ctrl[4], 1u) >= ctrl[1]) lab = -1;
    } else if (lab == 1) {
        const unsigned bin = hash_u32((unsigned)k, SEED_FG) >> 24;
        if (bin < ctrl[2]) lab = -1;
        else if (bin == ctrl[2] && atomicAdd(&ctrl[5], 1u) >= ctrl[3]) lab = -1;
    }
    labels[k] = lab;

    v4f t = (v4f){0.f, 0.f, 0.f, 0.f};
    if (lab == 1) {
        const v4f a  = __builtin_nontemporal_load((const v4f*)anchors + k);
        const v4f gb = ((const v4f*)s_gt)[arg_ov[k]];
        t.x = (gb.x - a.x) / a.z;
        t.y = (gb.y - a.y) / a.w;
        t.z = logf(gb.z / a.z);
        t.w = logf(gb.w / a.w);
    }
    // write-once stream: non-temporal b128 store
    __builtin_nontemporal_store(t, (v4f*)tgt + k);
}

// ---------------------------------------------------------------------------
extern "C" void kernel_launch(void* const* d_in, const int* in_sizes, int n_in,
                              void* d_out, int out_size, void* d_ws, size_t ws_size,
                              hipStream_t stream) {
    (void)in_sizes; (void)n_in; (void)out_size; (void)ws_size;

    // inputs: cls_scores (unused), gt_boxes (1,64,4) f32, anchors (1,15,128,128,4) f32
    const float* gt      = (const float*)d_in[1];
    const float* anchors = (const float*)d_in[2];

    // output: labels int32 [K] then targets f32 [K*4], concatenated flat
    int*   labels = (int*)d_out;
    float* tgt    = (float*)d_out + KANCH;

    // workspace layout
    char* ws = (char*)d_ws;
    unsigned long long* gt_pack = (unsigned long long*)(ws);            //  512 B
    unsigned*           hist    = (unsigned*)(ws + 512);                // 2048 B
    unsigned*           ctrl    = (unsigned*)(ws + 2560);               //   32 B
    int*                arg_ov  = (int*)(ws + 4096);                    // K*4 B
    unsigned char*      validb  = (unsigned char*)(ws + 4096 + (size_t)KANCH * 4);

    const int nblk = KANCH / 256;   // 960, exact

    k_init      <<<1,    512, 0, stream>>>(gt_pack, hist);
    k_iou       <<<nblk, 256, 0, stream>>>(anchors, gt, labels, arg_ov, validb,
                                           gt_pack, hist);
    k_fix_thresh<<<1,    256, 0, stream>>>(gt_pack, validb, labels, hist, ctrl);
    k_final     <<<nblk, 256, 0, stream>>>(anchors, gt, arg_ov, labels, tgt, ctrl);
}